// GraphBasedRelationNet_12249246728934
// MI455X (gfx1250) — compile-verified
//
#include <hip/hip_runtime.h>
#include <hip/hip_bf16.h>

#define DIM 128  // D == H == 128

typedef float v2f __attribute__((ext_vector_type(2)));
typedef float v8f __attribute__((ext_vector_type(8)));

// ---------------------------------------------------------------------------
// GEMM: C[r, :] = A[r, :] @ W + bias (+ residual), A: nrows x 128, W: 128x128.
// One wave (32 lanes) computes a 16-row x 128-col strip via 8 parallel
// 16x16 f32 accumulators, iterating K in steps of 4 with
// V_WMMA_F32_16X16X4_F32. 8 waves per 256-thread block.
//
// fp32 WMMA fragment layouts (ISA 7.12.2):
//   A (16x4):  lanes 0-15 -> M=lane,   VGPR0=K0, VGPR1=K1
//              lanes16-31 -> M=lane-16,VGPR0=K2, VGPR1=K3
//   B (4x16):  lanes 0-15 -> N=lane,   VGPR0=K0, VGPR1=K1
//              lanes16-31 -> N=lane-16,VGPR0=K2, VGPR1=K3
//   C (16x16): VGPR v: lanes 0-15 -> M=v, N=lane; lanes16-31 -> M=v+8
// ---------------------------------------------------------------------------
__global__ __launch_bounds__(256) void gemm_bias_kernel(
    const float* __restrict__ A, const float* __restrict__ W,
    const float* __restrict__ bias, const float* __restrict__ residual,
    float* __restrict__ C, int nrows)
{
  const int wave    = threadIdx.x >> 5;
  const int lane    = threadIdx.x & 31;
  const int rowTile = blockIdx.x * 8 + wave;
  if (rowTile * 16 >= nrows) return;  // wave-uniform: EXEC stays all-ones
  const int row0 = rowTile * 16;

  const int m  = lane & 15;          // M for A-frag, N for B-frag
  const int kb = (lane >> 4) << 1;   // K sub-offset: 0 for lanes 0-15, 2 for 16-31

  v8f acc[8];
#pragma unroll
  for (int t = 0; t < 8; ++t) acc[t] = (v8f){0.f,0.f,0.f,0.f,0.f,0.f,0.f,0.f};

  const float* Arow = A + (size_t)(row0 + m) * DIM;

  for (int k0 = 0; k0 < DIM; k0 += 4) {
    v2f a;
    a.x = Arow[k0 + kb + 0];
    a.y = Arow[k0 + kb + 1];
    const float* Wr0 = W + (size_t)(k0 + kb) * DIM;
    const float* Wr1 = Wr0 + DIM;
#pragma unroll
    for (int t = 0; t < 8; ++t) {
      const int col = t * 16 + m;
      v2f b;
      b.x = Wr0[col];
      b.y = Wr1[col];
      // (neg_a, A, neg_b, B, c_mod, C, reuse_a, reuse_b)
      acc[t] = __builtin_amdgcn_wmma_f32_16x16x4_f32(
          false, a, false, b, (short)0, acc[t], false, false);
    }
  }

  const int mhi = (lane >> 4) << 3;  // +8 row offset for lanes 16-31
#pragma unroll
  for (int t = 0; t < 8; ++t) {
    const int col = t * 16 + m;
    const float bv = bias[col];
#pragma unroll
    for (int v = 0; v < 8; ++v) {
      const int r = row0 + mhi + v;
      float val = acc[t][v] + bv;
      if (residual) val += residual[(size_t)r * DIM + col];
      C[(size_t)r * DIM + col] = val;
    }
  }
}

// ---------------------------------------------------------------------------
// Scatter-sum: one wave per edge; lane l owns features [4l, 4l+4) as float4.
// agg[dst] += xt[src]; deg[dst] += 1. Atomics resolve in L2 (agg is 5 MB,
// fully L2-resident on the 192 MB L2).
// ---------------------------------------------------------------------------
__global__ __launch_bounds__(256) void scatter_add_kernel(
    const float* __restrict__ xt, const long long* __restrict__ src,
    const long long* __restrict__ dst, float* __restrict__ agg,
    float* __restrict__ deg, int E)
{
  const long long gid = (long long)blockIdx.x * blockDim.x + threadIdx.x;
  const int e    = (int)(gid >> 5);
  const int lane = (int)(gid & 31);
  if (e >= E) return;
  const long long s = src[e];
  const long long d = dst[e];
  const float4 mv = ((const float4*)(xt + s * DIM))[lane];
  float* o = agg + d * DIM + (size_t)lane * 4;
  atomicAdd(o + 0, mv.x);
  atomicAdd(o + 1, mv.y);
  atomicAdd(o + 2, mv.z);
  atomicAdd(o + 3, mv.w);
  if (lane == 0) atomicAdd(deg + d, 1.0f);
}

// ---------------------------------------------------------------------------
// Fused: x = relu(LayerNorm(agg/deg (+ residual)) * g + be)
// One wave per node; float4 per lane; wave32 shuffle reductions.
// ---------------------------------------------------------------------------
__global__ __launch_bounds__(256) void ln_relu_kernel(
    const float* __restrict__ agg, const float* __restrict__ deg,
    const float* __restrict__ residual,
    const float* __restrict__ g, const float* __restrict__ be,
    float* __restrict__ out, int N)
{
  const int node = blockIdx.x * 8 + (threadIdx.x >> 5);
  const int lane = threadIdx.x & 31;
  if (node >= N) return;

  const float inv = 1.0f / fmaxf(deg[node], 1.0f);
  float4 v = ((const float4*)(agg + (size_t)node * DIM))[lane];
  v.x *= inv; v.y *= inv; v.z *= inv; v.w *= inv;
  if (residual) {
    const float4 r = ((const float4*)(residual + (size_t)node * DIM))[lane];
    v.x += r.x; v.y += r.y; v.z += r.z; v.w += r.w;
  }

  float s = v.x + v.y + v.z + v.w;
#pragma unroll
  for (int o = 16; o > 0; o >>= 1) s += __shfl_xor(s, o, 32);
  const float mu = s * (1.0f / DIM);

  const float d0 = v.x - mu, d1 = v.y - mu, d2 = v.z - mu, d3 = v.w - mu;
  float q = d0 * d0 + d1 * d1 + d2 * d2 + d3 * d3;
#pragma unroll
  for (int o = 16; o > 0; o >>= 1) q += __shfl_xor(q, o, 32);
  const float rstd = rsqrtf(q * (1.0f / DIM) + 1e-5f);

  const float4 gg = ((const float4*)g)[lane];
  const float4 bb = ((const float4*)be)[lane];
  float4 r;
  r.x = fmaxf(d0 * rstd * gg.x + bb.x, 0.0f);
  r.y = fmaxf(d1 * rstd * gg.y + bb.y, 0.0f);
  r.z = fmaxf(d2 * rstd * gg.z + bb.z, 0.0f);
  r.w = fmaxf(d3 * rstd * gg.w + bb.w, 0.0f);
  ((float4*)(out + (size_t)node * DIM))[lane] = r;
}

// ---------------------------------------------------------------------------
extern "C" void kernel_launch(void* const* d_in, const int* in_sizes, int n_in,
                              void* d_out, int out_size, void* d_ws, size_t ws_size,
                              hipStream_t stream) {
  const float*     x0  = (const float*)d_in[0];
  const long long* ei  = (const long long*)d_in[1];  // int64 (2, E)
  const float*     W1  = (const float*)d_in[2];
  const float*     b1  = (const float*)d_in[3];
  const float*     g1  = (const float*)d_in[4];
  const float*     be1 = (const float*)d_in[5];
  const float*     W2  = (const float*)d_in[6];
  const float*     b2  = (const float*)d_in[7];
  const float*     g2  = (const float*)d_in[8];
  const float*     be2 = (const float*)d_in[9];
  const float*     Wo  = (const float*)d_in[10];
  const float*     bo  = (const float*)d_in[11];
  float*           out = (float*)d_out;

  const int N = in_sizes[0] / DIM;   // 10000
  const int E = in_sizes[1] / 2;     // 640000
  const long long* src = ei;
  const long long* dst = ei + E;

  // workspace layout (floats)
  float* xt  = (float*)d_ws;                 // N*DIM  (post-GEMM messages)
  float* agg = xt  + (size_t)N * DIM;        // N*DIM  (scatter accumulator)
  float* x1  = agg + (size_t)N * DIM;        // N*DIM  (layer-0 activations)
  float* x2  = x1  + (size_t)N * DIM;        // N*DIM  (layer-1 activations)
  float* deg = x2  + (size_t)N * DIM;        // N      (in-degree)

  const dim3 blk(256);
  const int  gemm_grid    = ((N / 16) + 7) / 8;
  const int  scatter_grid = (int)(((long long)E * 32 + 255) / 256);
  const int  ln_grid      = (N + 7) / 8;

  // ---- layer 0: xt = x0 @ W1 + b1 ; scatter-mean ; x1 = relu(LN(.)) ----
  gemm_bias_kernel<<<gemm_grid, blk, 0, stream>>>(x0, W1, b1, nullptr, xt, N);
  hipMemsetAsync(agg, 0, (size_t)N * DIM * sizeof(float), stream);
  hipMemsetAsync(deg, 0, (size_t)N * sizeof(float), stream);
  scatter_add_kernel<<<scatter_grid, blk, 0, stream>>>(xt, src, dst, agg, deg, E);
  ln_relu_kernel<<<ln_grid, blk, 0, stream>>>(agg, deg, nullptr, g1, be1, x1, N);

  // ---- layer 1: xt = x1 @ W2 + b2 ; scatter-mean ; x2 = relu(LN(x1 + .)) ----
  gemm_bias_kernel<<<gemm_grid, blk, 0, stream>>>(x1, W2, b2, nullptr, xt, N);
  hipMemsetAsync(agg, 0, (size_t)N * DIM * sizeof(float), stream);
  hipMemsetAsync(deg, 0, (size_t)N * sizeof(float), stream);
  scatter_add_kernel<<<scatter_grid, blk, 0, stream>>>(xt, src, dst, agg, deg, E);
  ln_relu_kernel<<<ln_grid, blk, 0, stream>>>(agg, deg, x1, g2, be2, x2, N);

  // ---- output head: out = x2 @ Wo + bo + x0 ----
  gemm_bias_kernel<<<gemm_grid, blk, 0, stream>>>(x2, Wo, bo, x0, out, N);
}